// Decoder_23089744183379
// MI455X (gfx1250) — compile-verified
//
#include <hip/hip_runtime.h>
#include <hip/hip_bf16.h>
#include <math.h>

// Dimensions from the reference
#define HID 128
#define SEQ 256

// Workspace layout (floats)
#define WS_WHH   0            // 32 kc * 8 nt * 32 lanes * 2  = 16384 floats (Whh^T B-fragments)
#define WS_WOUT  16384        // 32 kc * 16 nt * 32 lanes * 2 = 32768 floats (Wout^T B-fragments)
#define WS_U     49152        // u[j]  = sum_k W_ih[j,k]*W_in[k]
#define WS_VB    49280        // v[j] + b_ih[j] + b_hh[j]
#define WS_C0    49408        // rowsum(W_ih)[j] + b_ih[j] + b_hh[j]   (t==0 input term)
#define WS_LOSS  49536        // 16 per-block loss partials

typedef float v2f __attribute__((ext_vector_type(2)));
typedef float v8f __attribute__((ext_vector_type(8)));

__device__ __forceinline__ v8f wmma4(v2f a, v2f b, v8f c) {
  // D(16x16,f32) = A(16x4,f32) * B(4x16,f32) + C
  return __builtin_amdgcn_wmma_f32_16x16x4_f32(false, a, false, b, (short)0, c, false, false);
}

// ---------------------------------------------------------------------------
// Setup: pack B-fragments for both GEMMs into d_ws, fold the rank-1 input
// projection into per-column vectors, zero the loss slot.
// B-fragment layout (mirrors 16x4 f32 A layout): for fragment (kc, nt),
// lane l holds float2 { B[4kc + 2*(l>>4)][16nt + (l&15)], B[4kc + 2*(l>>4)+1][...] }
// with B = W^T, i.e. B[k][n] = W[n][k].
// ---------------------------------------------------------------------------
__global__ void decoder_setup(const float* __restrict__ W_ih, const float* __restrict__ W_hh,
                              const float* __restrict__ b_ih, const float* __restrict__ b_hh,
                              const float* __restrict__ W_out, const float* __restrict__ W_in,
                              const float* __restrict__ b_in,
                              float* __restrict__ ws, float* __restrict__ out) {
  int tid = blockIdx.x * blockDim.x + threadIdx.x;
  if (tid == 0) out[0] = 0.0f;

  if (tid < 8192) {                      // Whh^T fragments: kc(32) x nt(8) x lane(32)
    int kc = tid >> 8;
    int rem = tid & 255;
    int nt = rem >> 5;
    int lane = rem & 31;
    int n  = nt * 16 + (lane & 15);
    int k0 = kc * 4 + ((lane >> 4) << 1);
    ws[WS_WHH + tid * 2 + 0] = W_hh[n * HID + k0];
    ws[WS_WHH + tid * 2 + 1] = W_hh[n * HID + k0 + 1];
  } else if (tid < 8192 + 16384) {       // Wout^T fragments: kc(32) x nt(16) x lane(32)
    int id = tid - 8192;
    int kc = id >> 9;
    int rem = id & 511;
    int nt = rem >> 5;
    int lane = rem & 31;
    int n  = nt * 16 + (lane & 15);      // n in [0,256)
    int k0 = kc * 4 + ((lane >> 4) << 1);
    ws[WS_WOUT + id * 2 + 0] = W_out[n * HID + k0];
    ws[WS_WOUT + id * 2 + 1] = W_out[n * HID + k0 + 1];
  } else if (tid < 8192 + 16384 + HID) { // folded input-projection vectors
    int j = tid - (8192 + 16384);
    float u = 0.0f, v = 0.0f, rs = 0.0f;
    for (int k = 0; k < HID; ++k) {
      float w = W_ih[j * HID + k];
      u  += w * W_in[k];                 // W_in is (H,1)
      v  += w * b_in[k];
      rs += w;
    }
    float bb = b_ih[j] + b_hh[j];
    ws[WS_U  + j] = u;
    ws[WS_VB + j] = v + bb;
    ws[WS_C0 + j] = rs + bb;             // dec_in = ones at t==0
  }
}

// ---------------------------------------------------------------------------
// Main: 16 blocks x 32 threads. Each wave owns 16 batch rows end-to-end:
// the full T=256 recurrence, both WMMA GEMMs, mask/softmax/argmax/CE.
// ---------------------------------------------------------------------------
__global__ __launch_bounds__(32) void decoder_main(const float* __restrict__ enc,
                                                   const int* __restrict__ targets,
                                                   const float* __restrict__ b_out,
                                                   float* __restrict__ ws,
                                                   float* __restrict__ out) {
  __shared__ __align__(16) float h_lds[16 * 132];  // 16 rows x 128 cols, stride 132 (bank-pad)
  __shared__ int tgc[16];                          // targets[b, t]
  __shared__ int tgp[16];                          // targets[b, t-1]
  __shared__ int chs[16];                          // chosen (prev) per row

  const int lane = threadIdx.x;
  const int lx = lane & 15;        // column-within-tile / row-for-A-frag
  const int hx = lane >> 4;        // half-wave index
  const int b0 = blockIdx.x * 16;  // first batch row of this wave

  // Per-lane column coefficients (column j = 16*nt + lx)
  float u_r[8], vb_r[8], c0_r[8];
#pragma unroll
  for (int nt = 0; nt < 8; ++nt) {
    int j = nt * 16 + lx;
    u_r[nt]  = ws[WS_U  + j];
    vb_r[nt] = ws[WS_VB + j];
    c0_r[nt] = ws[WS_C0 + j];
  }
  float bo_r[16];
#pragma unroll
  for (int nt = 0; nt < 16; ++nt) bo_r[nt] = b_out[nt * 16 + lx];

  // h0 = encoder_context[0, b0:b0+16, :]
  for (int i = lane; i < 16 * HID; i += 32)
    h_lds[(i >> 7) * 132 + (i & 127)] = enc[b0 * HID + i];

  // mask bits: bit nt of mbits[r] == masked(row r+8*hx, column 16*nt+lx)
  unsigned mbits[8];
#pragma unroll
  for (int r = 0; r < 8; ++r) mbits[r] = 0u;

  float lossAcc = 0.0f;
  const v8f vzero = {};
  __syncthreads();

  for (int t = 0; t < SEQ; ++t) {
    if (lane < 16)      tgc[lane]      = targets[(b0 + lane) * SEQ + t];
    else if (t > 0)     tgp[lane - 16] = targets[(b0 + lane - 16) * SEQ + t - 1];
    __syncthreads();

    // Reference: mask update happens only for t>0 (upd_mask with prev chosen)
    if (t > 0) {
#pragma unroll
      for (int r = 0; r < 8; ++r) {
        int c = chs[r + 8 * hx];
        if ((c & 15) == lx) mbits[r] |= 1u << (c >> 4);
      }
    }

    // ---------------- GEMM1: pre = h @ Whh^T (f32 WMMA) ----------------
    v8f acc1[8];
#pragma unroll
    for (int nt = 0; nt < 8; ++nt) acc1[nt] = vzero;
    for (int kc = 0; kc < 32; ++kc) {
      v2f a = *(const v2f*)(h_lds + lx * 132 + kc * 4 + hx * 2);  // A frag: M=lx, K chunk
#pragma unroll
      for (int nt = 0; nt < 8; ++nt) {
        v2f b = *(const v2f*)(ws + WS_WHH + ((kc * 8 + nt) * 32 + lane) * 2);
        acc1[nt] = wmma4(a, b, acc1[nt]);
      }
    }
    __syncthreads();

    // epilogue1: + (tgt_{t-1}*u + v + b) [or c0 at t==0], tanh, write new h
#pragma unroll
    for (int nt = 0; nt < 8; ++nt) {
      int j = nt * 16 + lx;
#pragma unroll
      for (int r = 0; r < 8; ++r) {
        int row = r + 8 * hx;
        float cin = (t == 0) ? c0_r[nt]
                             : ((float)tgp[row]) * u_r[nt] + vb_r[nt];
        h_lds[row * 132 + j] = tanhf(acc1[nt][r] + cin);
      }
    }
    __syncthreads();

    if (t == 0) {
      // masked_0 is a constant one-hot: logits unused. sm = onehot(0).
      if (lane < 16) { chs[lane] = 0; out[1 + (b0 + lane) * SEQ] = 0.0f; }
      if (lx == 0) {
        float S2 = expf(1.0f) + 255.0f;   // sum_j exp(sm_j)
#pragma unroll
        for (int r = 0; r < 8; ++r) {
          int row = r + 8 * hx;
          float st = (tgc[row] == 0) ? 1.0f : 0.0f;
          lossAcc += logf(S2) - st;       // ce = log(S2) - sm[tgt]
        }
      }
      __syncthreads();
      continue;
    }

    // ---------------- GEMM2: logits = h_new @ Wout^T ----------------
    v8f acc2[16];
#pragma unroll
    for (int nt = 0; nt < 16; ++nt) acc2[nt] = vzero;
    for (int kc = 0; kc < 32; ++kc) {
      v2f a = *(const v2f*)(h_lds + lx * 132 + kc * 4 + hx * 2);
#pragma unroll
      for (int nt = 0; nt < 16; ++nt) {
        v2f b = *(const v2f*)(ws + WS_WOUT + ((kc * 16 + nt) * 32 + lane) * 2);
        acc2[nt] = wmma4(a, b, acc2[nt]);
      }
    }

    // epilogue2: per-row masked softmax stats, argmax, CE (shfl over 16-lane halves)
#pragma unroll
    for (int r = 0; r < 8; ++r) {
      int row = r + 8 * hx;
      unsigned mb = mbits[r];

      // pass 1: biased logits (cached) + masked max/argmax
      float lv[16];
      float m = -INFINITY; int ai = 0x7fffffff;
#pragma unroll
      for (int nt = 0; nt < 16; ++nt) {
        float vv = acc2[nt][r] + bo_r[nt];
        lv[nt] = vv;
        bool msk = (mb >> nt) & 1u;
        if (!msk && vv > m) { m = vv; ai = nt * 16 + lx; }   // first-index tie-break per lane
      }
#pragma unroll
      for (int s = 1; s < 16; s <<= 1) {
        float om = __shfl_xor(m, s, 32);
        int   oi = __shfl_xor(ai, s, 32);
        if (om > m || (om == m && oi < ai)) { m = om; ai = oi; }
      }

      // pass 2: e = exp(l - m) (0 if masked), computed ONCE; Z = sum e
      float e[16];
      float z = 0.0f;
#pragma unroll
      for (int nt = 0; nt < 16; ++nt) {
        bool msk = (mb >> nt) & 1u;
        float ev = msk ? 0.0f : expf(lv[nt] - m);
        e[nt] = ev;
        z += ev;
      }
#pragma unroll
      for (int s = 1; s < 16; s <<= 1) z += __shfl_xor(z, s, 32);
      float rz = 1.0f / z;                 // one reciprocal per row (vs 16 divisions)

      // pass 3: sm = e * rz; S2 = sum exp(sm) (masked give exp(0)=1); st = sm[tgt]
      float s2 = 0.0f, st = 0.0f;
      int tcur = tgc[row];
#pragma unroll
      for (int nt = 0; nt < 16; ++nt) {
        float sv = e[nt] * rz;
        s2 += expf(sv);
        st += (nt * 16 + lx == tcur) ? sv : 0.0f;
      }
#pragma unroll
      for (int s = 1; s < 16; s <<= 1) {
        s2 += __shfl_xor(s2, s, 32);
        st += __shfl_xor(st, s, 32);
      }

      if (lx == 0) {
        lossAcc += logf(s2) - st;        // ce = -(sm_tgt - log sum exp(sm))
        chs[row] = ai;
        out[1 + (b0 + row) * SEQ + t] = (float)ai;
      }
    }
    __syncthreads();
  }

  // combine the two half-wave loss accumulators (lanes 0 and 16), write partial
  lossAcc += __shfl_xor(lossAcc, 16, 32);
  if (lane == 0) ws[WS_LOSS + blockIdx.x] = lossAcc;
}

// Deterministic fixed-order loss reduction: loss = sum_{b,t} ce / B^2
__global__ void decoder_finalize(const float* __restrict__ ws, float* __restrict__ out) {
  if (blockIdx.x == 0 && threadIdx.x == 0) {
    float s = 0.0f;
    for (int i = 0; i < 16; ++i) s += ws[WS_LOSS + i];
    out[0] = s * (1.0f / 65536.0f);
  }
}

extern "C" void kernel_launch(void* const* d_in, const int* in_sizes, int n_in,
                              void* d_out, int out_size, void* d_ws, size_t ws_size,
                              hipStream_t stream) {
  const float* enc     = (const float*)d_in[0];   // (1,256,128) f32
  const int*   targets = (const int*)  d_in[1];   // (256,256)   i32
  const float* W_ih    = (const float*)d_in[2];   // (128,128)
  const float* W_hh    = (const float*)d_in[3];   // (128,128)
  const float* b_ih    = (const float*)d_in[4];   // (128,)
  const float* b_hh    = (const float*)d_in[5];   // (128,)
  const float* W_out   = (const float*)d_in[6];   // (256,128)
  const float* b_out   = (const float*)d_in[7];   // (256,)
  const float* W_in    = (const float*)d_in[8];   // (128,1)
  const float* b_in    = (const float*)d_in[9];   // (128,)
  float* out = (float*)d_out;                      // [0]=loss, [1..]=tours (256,1,256)
  float* ws  = (float*)d_ws;

  // 8192 + 16384 + 128 = 24704 setup work-items
  decoder_setup<<<97, 256, 0, stream>>>(W_ih, W_hh, b_ih, b_hh, W_out, W_in, b_in, ws, out);
  // 16 waves, one 16-row batch tile each, on separate WGPs
  decoder_main<<<16, 32, 0, stream>>>(enc, targets, b_out, ws, out);
  decoder_finalize<<<1, 32, 0, stream>>>(ws, out);
}